// GraphLayer_18777597018206
// MI455X (gfx1250) — compile-verified
//
#include <hip/hip_runtime.h>

// Fused GraphLayer for MI455X (gfx1250):
//   h_out = einsum('fde,bfe->bfd', W_out, h)      step 1 (per-field GEMM, WMMA)
//   aggr  = einsum('bfg,bgd->bfd', g, h_out)      step 2 (per-batch GEMM, WMMA)
//   a     = einsum('fde,bfe->bfd', W_in, aggr)+b  step 3 (per-field GEMM, WMMA)
// Single kernel, intermediates in LDS (~273 MB total HBM traffic).
// g[b] staging uses the Tensor Data Mover (TENSOR_LOAD_TO_LDS), double-buffered
// so the DMA overlaps the WMMA compute; first batch's DMA overlaps step 1.

#define B_TOT 16384
#define F_DIM 40
#define D_DIM 32
#define F2    (F_DIM * F_DIM)

typedef __attribute__((ext_vector_type(2))) float v2f;
typedef __attribute__((ext_vector_type(8))) float v8f;
typedef unsigned int u32x4 __attribute__((ext_vector_type(4)));
typedef int          i32x4 __attribute__((ext_vector_type(4)));
typedef int          i32x8 __attribute__((ext_vector_type(8)));

#if defined(__has_builtin)
#  if __has_builtin(__builtin_amdgcn_tensor_load_to_lds) && __has_builtin(__builtin_amdgcn_s_wait_tensorcnt)
#    define HAVE_TDM 1
#  else
#    define HAVE_TDM 0
#  endif
#else
#  define HAVE_TDM 0
#endif

static __device__ __forceinline__ v8f wmma4(v2f a, v2f b, v8f c) {
  // D = A(16x4 f32) * B(4x16 f32) + C(16x16 f32)
  return __builtin_amdgcn_wmma_f32_16x16x4_f32(
      /*neg_a=*/false, a, /*neg_b=*/false, b,
      /*c_mod=*/(short)0, c, /*reuse_a=*/false, /*reuse_b=*/false);
}

constexpr int BT = 16;        // batches per workgroup (WMMA M dimension)
constexpr int NW = 4;         // waves per workgroup
constexpr int HSTRIDE = 1284; // LDS dwords per batch (1280 + pad: stride%64==4 -> conflict-free A frags)

#if HAVE_TDM
// 1-D TDM descriptor: copy 1600 contiguous f32 (g[b], 6.4 KB) from global -> LDS.
static __device__ __forceinline__ void tdm_load_g(const float* gsrc, unsigned lds_byte_addr) {
  unsigned long long ga = (unsigned long long)(size_t)gsrc;
  u32x4 g0;
  g0.x = 1u;                                           // count=1, user descriptor, no gather
  g0.y = lds_byte_addr;                                // lds_addr  (bits 63:32)
  g0.z = (unsigned)(ga & 0xFFFFFFFFu);                 // global_addr[31:0]
  g0.w = (unsigned)((ga >> 32) & 0x01FFFFFFu)          // global_addr[56:32]
       | 0x80000000u;                                  // type=2 ("image") in bits 127:126
  i32x8 g1;
  g1[0] = 0x00020000;                                  // workgroup_mask=0, data_size=2 (4 bytes)
  g1[1] = (int)(1600u << 16);                          // tensor_dim0 = 1600 (bits 79:48)
  g1[2] = (int)(1u << 16);                             // tensor_dim1 = 1   (bits 111:80)
  g1[3] = (int)(1600u << 16);                          // tile_dim0  = 1600 (bits 127:112)
  g1[4] = 1;                                           // tile_dim1 = 1, tile_dim2 = 0
  g1[5] = 1600;                                        // tensor_dim0_stride (bits 207:160)
  g1[6] = 0;
  g1[7] = 0;
  i32x4 z4 = {0, 0, 0, 0};
#if __clang_major__ >= 23
  i32x8 z8 = {0, 0, 0, 0, 0, 0, 0, 0};
  __builtin_amdgcn_tensor_load_to_lds(g0, g1, z4, z4, z8, 0);
#else
  __builtin_amdgcn_tensor_load_to_lds(g0, g1, z4, z4, 0);
#endif
}
static __device__ __forceinline__ void tdm_wait0() {
  asm volatile("" ::: "memory");
  __builtin_amdgcn_s_wait_tensorcnt(0);
  asm volatile("" ::: "memory");
}
#endif

__global__ __launch_bounds__(NW * 32)
void fused_graph_layer(const float* __restrict__ g,
                       const float* __restrict__ h,
                       const float* __restrict__ Win,
                       const float* __restrict__ Wout,
                       const float* __restrict__ bias,
                       float* __restrict__ out)
{
  __shared__ float s_hout[BT * HSTRIDE];        // h_out, later overwritten by aggr (~82 KB)
  __shared__ float s_g[NW * 2 * F2];            // per-wave double-buffered g[b] staging (~51 KB)

  const int tid  = threadIdx.x;
  const int w    = tid >> 5;     // wave id 0..3
  const int lane = tid & 31;
  const int lo   = lane & 15;    // lane within half-wave
  const int hi   = lane >> 4;    // half-wave select (K split for A/B frags)
  const long b0  = (long)blockIdx.x * BT;
  const int blFirst = w * (BT / NW);
  const int blEnd   = blFirst + (BT / NW);

  const v8f Z8 = {0.f,0.f,0.f,0.f,0.f,0.f,0.f,0.f};

#if HAVE_TDM
  // Kick off the DMA for this wave's first g[b] immediately: it overlaps all of step 1.
  tdm_load_g(g + (b0 + blFirst) * (long)F2,
             (unsigned)(size_t)(&s_g[(w * 2 + 0) * F2]));
#endif

  // ---------------- Step 1: h_out[16 x 32] = h[16 x 32] * W_out[f]^T per field ----------------
  for (int f = w; f < F_DIM; f += NW) {
    v8f c0 = Z8, c1 = Z8;
    const float* hA = h + (b0 + lo) * (long)(F_DIM * D_DIM) + f * D_DIM; // A: M=batch(lane), K=e
    const float* wB = Wout + (long)f * (D_DIM * D_DIM);                  // W[f][d][e]
#pragma unroll
    for (int k = 0; k < 8; ++k) {
      const int e0 = 4 * k + 2 * hi;
      v2f a  = *(const v2f*)(hA + e0);                    // (M=lo, K=e0..e0+1)
      v2f bL = *(const v2f*)(wB + lo * D_DIM + e0);       // (K=e0..e0+1, N=d=lo)
      v2f bH = *(const v2f*)(wB + (16 + lo) * D_DIM + e0);// (K, N=d=16+lo)
      c0 = wmma4(a, bL, c0);
      c1 = wmma4(a, bH, c1);
    }
    // C/D layout: VGPR r, lane l -> (M = r + 8*hi, N = lo)
    float* dst = s_hout + (8 * hi) * HSTRIDE + f * D_DIM + lo;
#pragma unroll
    for (int r = 0; r < 8; ++r) {
      dst[r * HSTRIDE]      = c0[r];
      dst[r * HSTRIDE + 16] = c1[r];
    }
  }
  __syncthreads();

  // ---------------- Step 2: aggr[b] = g[b](40x40) * h_out[b](40x32), one batch at a time ----------------
  for (int bl = blFirst; bl < blEnd; ++bl) {
    const int buf = (bl - blFirst) & 1;
    const float* gw = s_g + (w * 2 + buf) * F2;

#if HAVE_TDM
    tdm_wait0();                                          // current buffer's DMA complete
    if (bl + 1 < blEnd)                                   // prefetch next batch, overlaps compute
      tdm_load_g(g + (b0 + bl + 1) * (long)F2,
                 (unsigned)(size_t)(&s_g[(w * 2 + (buf ^ 1)) * F2]));
#else
    const float* gsrc = g + (b0 + bl) * (long)F2;
    float* gwm = s_g + (w * 2 + buf) * F2;
#pragma unroll
    for (int i = 0; i < F2 / 32; ++i)                     // coalesced staging of g[b] (1600 f32)
      gwm[i * 32 + lane] = gsrc[i * 32 + lane];
#endif

    const float* hb = s_hout + bl * HSTRIDE;              // B matrix: h_out[b][field][d]
    v8f acc[3][2];
#pragma unroll
    for (int mt = 0; mt < 3; ++mt) { acc[mt][0] = Z8; acc[mt][1] = Z8; }

    for (int k = 0; k < 10; ++k) {                        // K = field dimension, 40 = 10*4 exact
      const int kk = 4 * k + 2 * hi;
      v2f bL = { hb[kk * D_DIM + lo],       hb[(kk + 1) * D_DIM + lo] };
      v2f bH = { hb[kk * D_DIM + 16 + lo],  hb[(kk + 1) * D_DIM + 16 + lo] };
#pragma unroll
      for (int mt = 0; mt < 3; ++mt) {
        int row = mt * 16 + lo;                           // output field row (M), pad 40->48
        if (row >= F_DIM) row = F_DIM - 1;                // clamp: garbage masked at store
        v2f a = *(const v2f*)(gw + row * F_DIM + kk);
        acc[mt][0] = wmma4(a, bL, acc[mt][0]);
        acc[mt][1] = wmma4(a, bH, acc[mt][1]);
      }
    }
    float* dst = s_hout + bl * HSTRIDE;                   // overwrite h_out[b] with aggr[b]
#pragma unroll
    for (int mt = 0; mt < 3; ++mt) {
#pragma unroll
      for (int r = 0; r < 8; ++r) {
        const int fld = mt * 16 + r + 8 * hi;
        if (fld < F_DIM) {
          dst[fld * D_DIM + lo]      = acc[mt][0][r];
          dst[fld * D_DIM + 16 + lo] = acc[mt][1][r];
        }
      }
    }
  }
  __syncthreads();

  // ---------------- Step 3: out[16 x 32] = aggr[16 x 32] * W_in[f]^T + bias, per field ----------------
  const float bias0 = bias[lo];
  const float bias1 = bias[16 + lo];
  for (int f = w; f < F_DIM; f += NW) {
    v8f c0 = Z8, c1 = Z8;
    const float* aA = s_hout + lo * HSTRIDE + f * D_DIM;  // A from LDS: M=batch, K=e (bank-conflict-free)
    const float* wB = Win + (long)f * (D_DIM * D_DIM);
#pragma unroll
    for (int k = 0; k < 8; ++k) {
      const int e0 = 4 * k + 2 * hi;
      v2f a  = *(const v2f*)(aA + e0);
      v2f bL = *(const v2f*)(wB + lo * D_DIM + e0);
      v2f bH = *(const v2f*)(wB + (16 + lo) * D_DIM + e0);
      c0 = wmma4(a, bL, c0);
      c1 = wmma4(a, bH, c1);
    }
    float* dst = out + (b0 + 8 * hi) * (long)(F_DIM * D_DIM) + f * D_DIM + lo;
#pragma unroll
    for (int r = 0; r < 8; ++r) {
      dst[(long)r * (F_DIM * D_DIM)]      = c0[r] + bias0;
      dst[(long)r * (F_DIM * D_DIM) + 16] = c1[r] + bias1;
    }
  }
}

extern "C" void kernel_launch(void* const* d_in, const int* in_sizes, int n_in,
                              void* d_out, int out_size, void* d_ws, size_t ws_size,
                              hipStream_t stream) {
  // setup_inputs order: g, h, W_in, W_out, bias_p
  const float* g    = (const float*)d_in[0];
  const float* h    = (const float*)d_in[1];
  const float* Win  = (const float*)d_in[2];
  const float* Wout = (const float*)d_in[3];
  const float* bias = (const float*)d_in[4];
  float* out = (float*)d_out;

  dim3 grid(B_TOT / BT);
  dim3 block(NW * 32);
  hipLaunchKernelGGL(fused_graph_layer, grid, block, 0, stream,
                     g, h, Win, Wout, bias, out);
}